// MemEffAttention_42494406426865
// MI455X (gfx1250) — compile-verified
//
#include <hip/hip_runtime.h>
#include <cmath>

#define B_ 4
#define N_ 1025
#define C_ 768
#define H_ 12
#define DH_ 64
#define NT_ 1024            // body tokens per batch
#define QKV_COLS 2304       // 3*C
#define M_ROWS (B_*N_)      // 4100

#define GEMM_BN 256         // block N tile (4 waves x 64)
#define GEMM_KC 32          // K chunk staged in LDS
#define LDS_STRIDE 36       // padded fp32 stride: 16B aligned rows, 16 distinct banks

typedef float v2f __attribute__((ext_vector_type(2)));
typedef float v8f __attribute__((ext_vector_type(8)));

static __device__ __forceinline__ v8f wmma_f32_16x16x4(v2f a, v2f b, v8f c) {
  // D = A(16x4 f32) * B(4x16 f32) + C(16x16 f32), wave32
  return __builtin_amdgcn_wmma_f32_16x16x4_f32(false, a, false, b, (short)0, c, false, false);
}

typedef __attribute__((address_space(3))) const void lds_cv_t;
static __device__ __forceinline__ unsigned lds_byte_offset(const void* p) {
  return (unsigned)(uintptr_t)(lds_cv_t*)p;
}
// GLOBAL_LOAD_ASYNC_TO_LDS_B128: per-lane 16B gather from global VA -> LDS offset.
// Tracked by ASYNCcnt (CDNA5 ISA / CDNA5 ISA opcode 98).
static __device__ __forceinline__ void async_copy16(unsigned ldsoff, const void* gptr) {
  asm volatile("global_load_async_to_lds_b128 %0, %1, off"
               :: "v"(ldsoff), "v"((unsigned long long)(uintptr_t)gptr)
               : "memory");
}

static __device__ __forceinline__ float grp16_max(float v) {
  v = fmaxf(v, __shfl_xor(v, 1, 32));
  v = fmaxf(v, __shfl_xor(v, 2, 32));
  v = fmaxf(v, __shfl_xor(v, 4, 32));
  v = fmaxf(v, __shfl_xor(v, 8, 32));
  return v;
}
static __device__ __forceinline__ float grp16_sum(float v) {
  v += __shfl_xor(v, 1, 32);
  v += __shfl_xor(v, 2, 32);
  v += __shfl_xor(v, 4, 32);
  v += __shfl_xor(v, 8, 32);
  return v;
}
static __device__ __forceinline__ float wave_sum(float v) {
  v += __shfl_xor(v, 16, 32);
  return grp16_sum(v);
}

// ---------------------------------------------------------------------------
// fp32 WMMA GEMM: D[M,Nn] = A[M,K] * B[K,Nn] (+ bias[Nn]).
// Block = 128 threads = 4 waves; block tile 16(M) x 256(N), wave tile 16x64.
// A K-chunks (16x32) double-buffered in LDS via global_load_async_to_lds_b128.
// Requires Nn % 256 == 0, K % 32 == 0.
// ---------------------------------------------------------------------------
__global__ __launch_bounds__(128) void gemm_async_kernel(
    const float* __restrict__ A, const float* __restrict__ Bm,
    const float* __restrict__ bias, float* __restrict__ D,
    int M, int Nn, int K)
{
  __shared__ float bufA[2][16 * LDS_STRIDE];
  const int t    = threadIdx.x;
  const int lane = t & 31;
  const int w    = t >> 5;                       // wave 0..3
  const int row0 = blockIdx.x << 4;
  const int coln0 = blockIdx.y * GEMM_BN + w * 64;
  const int n16   = lane & 15;
  const int khalf = (lane >> 4) << 1;            // 0 or 2
  const int hi8   = (lane >> 4) << 3;

  // staging: thread t fetches A[row0 + t/8][kc + (t%8)*4 .. +3] (16B, aligned)
  const int srow = t >> 3;                       // 0..15
  const int skq  = (t & 7) << 2;                 // 0,4,...,28
  int srowg = row0 + srow; if (srowg > M - 1) srowg = M - 1;
  const float* agp = A + (size_t)srowg * K + skq;
  unsigned ldsoff0 = lds_byte_offset(&bufA[0][srow * LDS_STRIDE + skq]);
  unsigned ldsoff1 = lds_byte_offset(&bufA[1][srow * LDS_STRIDE + skq]);

  const v8f zero8 = {0.f,0.f,0.f,0.f,0.f,0.f,0.f,0.f};
  v8f acc[4];
  #pragma unroll
  for (int ct = 0; ct < 4; ++ct) acc[ct] = zero8;

  const int nchunk = K / GEMM_KC;
  async_copy16(ldsoff0, (const void*)agp);       // prologue: chunk 0 -> buf0
  for (int c = 0; c < nchunk; ++c) {
    if (c + 1 < nchunk) {
      async_copy16(((c + 1) & 1) ? ldsoff1 : ldsoff0,
                   (const void*)(agp + (size_t)(c + 1) * GEMM_KC));
      asm volatile("s_wait_asynccnt 0x1" ::: "memory");   // chunk c landed
    } else {
      asm volatile("s_wait_asynccnt 0x0" ::: "memory");
    }
    __syncthreads();                             // all waves' chunk-c parts visible
    const float* __restrict__ ab = &bufA[c & 1][0];
    const int kc = c * GEMM_KC;
    #pragma unroll
    for (int ks = 0; ks < 8; ++ks) {
      const int kl = ks * 4 + khalf;
      v2f a; a.x = ab[n16 * LDS_STRIDE + kl]; a.y = ab[n16 * LDS_STRIDE + kl + 1];
      const float* bp = Bm + (size_t)(kc + kl) * Nn + coln0 + n16;
      #pragma unroll
      for (int ct = 0; ct < 4; ++ct) {
        v2f b; b.x = bp[ct * 16]; b.y = bp[Nn + ct * 16];
        acc[ct] = wmma_f32_16x16x4(a, b, acc[ct]);
      }
    }
    __syncthreads();                             // buffer reusable next round
  }
  #pragma unroll
  for (int ct = 0; ct < 4; ++ct) {
    const int col = coln0 + ct * 16 + n16;
    const float badd = bias ? bias[col] : 0.f;
    #pragma unroll
    for (int r = 0; r < 8; ++r) {
      int row = row0 + r + hi8;
      if (row < M) D[(size_t)row * Nn + col] = acc[ct][r] + badd;
    }
  }
}

// ---------------------------------------------------------------------------
// Per-token mean-over-heads cosine similarity with q_g = g_info[0,0,:].
// One wave per (b, token); token rows are n = 1..1024.
// ---------------------------------------------------------------------------
__global__ __launch_bounds__(256) void qsim_kernel(
    const float* __restrict__ qkv, const float* __restrict__ g_info,
    float* __restrict__ qsim)
{
  const int w = (blockIdx.x * blockDim.x + threadIdx.x) >> 5;
  const int lane = threadIdx.x & 31;
  if (w >= B_ * NT_) return;              // uniform per wave
  const int b = w >> 10;
  const int t = w & 1023;
  const float* __restrict__ qp = qkv + (size_t)(b * N_ + t + 1) * QKV_COLS;
  float sim = 0.f;
  for (int h = 0; h < H_; ++h) {
    const float* q = qp + h * DH_;
    const float* g = g_info + h * DH_;
    float x0 = q[lane], x1 = q[lane + 32];
    float g0 = g[lane], g1 = g[lane + 32];
    float dot = wave_sum(x0 * g0 + x1 * g1);
    float qq  = wave_sum(x0 * x0 + x1 * x1);
    float gg  = wave_sum(g0 * g0 + g1 * g1);
    sim += dot / (sqrtf(qq) * sqrtf(gg));
  }
  if (lane == 0) qsim[w] = sim / 12.0f;
}

// ---------------------------------------------------------------------------
// Global min/max over q_sim, threshold 0.9 -> per-(b,token) masks + keep flags.
// Single block of 1024 threads (one thread per token).
// ---------------------------------------------------------------------------
__global__ __launch_bounds__(1024) void mask_kernel(
    const float* __restrict__ qsim,
    float* __restrict__ mpos, float* __restrict__ mneg,
    float* __restrict__ keepp, float* __restrict__ keepn)
{
  __shared__ float smin[1024];
  __shared__ float smax[1024];
  const int t = threadIdx.x;
  float lmin = 3.4e38f, lmax = -3.4e38f;
  for (int i = t; i < B_ * NT_; i += 1024) {
    float v = qsim[i];
    lmin = fminf(lmin, v); lmax = fmaxf(lmax, v);
  }
  smin[t] = lmin; smax[t] = lmax;
  __syncthreads();
  for (int s = 512; s > 0; s >>= 1) {
    if (t < s) {
      smin[t] = fminf(smin[t], smin[t + s]);
      smax[t] = fmaxf(smax[t], smax[t + s]);
    }
    __syncthreads();
  }
  const float gmin = smin[0], gmax = smax[0];
  bool anyp = false, anyn = false;
  for (int b = 0; b < B_; ++b) {
    float norm = (qsim[b * NT_ + t] - gmin) / (gmax - gmin);
    bool p = norm > 0.9f;
    mpos[b * NT_ + t] = p ? 1.f : 0.f;
    mneg[b * NT_ + t] = p ? 0.f : 1.f;
    anyp |= p; anyn |= !p;
  }
  keepp[t] = anyp ? 1.f : 0.f;
  keepn[t] = anyn ? 1.f : 0.f;
}

// ---------------------------------------------------------------------------
// Masked group attention, flash-style online softmax.
// grid = (65 query tiles, B*H, 2 groups), block = one wave (32).
// q/k/v body rows scaled by per-(b,token) mask (CLS row unmasked);
// keys masked to -inf by the any-batch keep flag. All WMMA fp32.
// ---------------------------------------------------------------------------
__global__ __launch_bounds__(32) void attn_kernel(
    const float* __restrict__ qkv,
    const float* __restrict__ mpos, const float* __restrict__ mneg,
    const float* __restrict__ keepp, const float* __restrict__ keepn,
    float* __restrict__ outp, float* __restrict__ outn)
{
  const int lane = threadIdx.x;
  const int qt = blockIdx.x;                 // 0..64
  const int b  = blockIdx.y / H_;
  const int h  = blockIdx.y % H_;
  const int g  = blockIdx.z;
  const float* __restrict__ mg = g ? mneg : mpos;
  const float* __restrict__ kk_keep = g ? keepn : keepp;
  float* __restrict__ outb = g ? outn : outp;
  const float scale = 0.125f;                // 64^-0.5
  __shared__ float ldsP[16 * 16];
  const int n16   = lane & 15;
  const int khalf = (lane >> 4) << 1;
  const int hi8   = (lane >> 4) << 3;
  const v8f zero8 = {0.f,0.f,0.f,0.f,0.f,0.f,0.f,0.f};

  // Q fragments (A-layout), 16 k-steps covering Dh=64
  v2f qa[16];
  {
    int row  = qt * 16 + n16;
    int rowc = row > 1024 ? 1024 : row;
    const float* qp = qkv + (size_t)(b * N_ + rowc) * QKV_COLS + h * DH_;
    float mq = (row == 0) ? 1.f : ((row <= 1024) ? mg[b * NT_ + row - 1] : 0.f);
    #pragma unroll
    for (int kk = 0; kk < 16; ++kk) {
      int k0 = kk * 4 + khalf;
      v2f a; a.x = qp[k0] * mq; a.y = qp[k0 + 1] * mq;
      qa[kk] = a;
    }
  }

  v8f acc[4];
  float mrun[8], lrun[8];
  #pragma unroll
  for (int ct = 0; ct < 4; ++ct) acc[ct] = zero8;
  #pragma unroll
  for (int r = 0; r < 8; ++r) { mrun[r] = -__builtin_inff(); lrun[r] = 0.f; }

  for (int kt = 0; kt < 65; ++kt) {
    const int key0 = kt * 16;
    // S = Q * K^T (16x16)
    v8f s = zero8;
    {
      int krow = key0 + n16;
      int krc  = krow > 1024 ? 1024 : krow;
      const float* kp = qkv + (size_t)(b * N_ + krc) * QKV_COLS + C_ + h * DH_;
      float mk = (krow == 0) ? 1.f : ((krow <= 1024) ? mg[b * NT_ + krow - 1] : 0.f);
      #pragma unroll
      for (int kk = 0; kk < 16; ++kk) {
        int k0 = kk * 4 + khalf;
        v2f bb; bb.x = kp[k0] * mk; bb.y = kp[k0 + 1] * mk;
        s = wmma_f32_16x16x4(qa[kk], bb, s);
      }
    }
    // online softmax update (rows live per 16-lane group in C-layout)
    const int key = key0 + n16;
    const bool kept = (key == 0) || (key <= 1024 && kk_keep[key - 1] > 0.5f);
    float p[8];
    #pragma unroll
    for (int r = 0; r < 8; ++r) {
      float sv   = kept ? s[r] * scale : -__builtin_inff();
      float tmax = grp16_max(sv);
      float mnew = fmaxf(mrun[r], tmax);
      float corr = expf(mrun[r] - mnew);
      float pv   = expf(sv - mnew);
      float tsum = grp16_sum(pv);
      lrun[r] = lrun[r] * corr + tsum;
      mrun[r] = mnew;
      p[r] = pv;
      #pragma unroll
      for (int ct = 0; ct < 4; ++ct) acc[ct][r] *= corr;
    }
    // P: C-layout -> LDS matrix -> A-layout fragments
    __syncthreads();
    #pragma unroll
    for (int r = 0; r < 8; ++r) ldsP[(r + hi8) * 16 + n16] = p[r];
    __syncthreads();
    v2f af[4];
    #pragma unroll
    for (int ks = 0; ks < 4; ++ks) {
      int k0 = ks * 4 + khalf;
      v2f a; a.x = ldsP[n16 * 16 + k0]; a.y = ldsP[n16 * 16 + k0 + 1];
      af[ks] = a;
    }
    // O += P * V  (16x64 as 4 column tiles)
    #pragma unroll
    for (int ks = 0; ks < 4; ++ks) {
      int k0 = key0 + ks * 4 + khalf;
      int k1 = k0 + 1;
      int k0c = k0 > 1024 ? 1024 : k0;
      int k1c = k1 > 1024 ? 1024 : k1;
      const float* vp0 = qkv + (size_t)(b * N_ + k0c) * QKV_COLS + 2 * C_ + h * DH_;
      const float* vp1 = qkv + (size_t)(b * N_ + k1c) * QKV_COLS + 2 * C_ + h * DH_;
      float mv0 = (k0 == 0) ? 1.f : ((k0 <= 1024) ? mg[b * NT_ + k0 - 1] : 0.f);
      float mv1 = (k1 == 0) ? 1.f : ((k1 <= 1024) ? mg[b * NT_ + k1 - 1] : 0.f);
      #pragma unroll
      for (int ct = 0; ct < 4; ++ct) {
        v2f bb;
        bb.x = vp0[ct * 16 + n16] * mv0;
        bb.y = vp1[ct * 16 + n16] * mv1;
        acc[ct] = wmma_f32_16x16x4(af[ks], bb, acc[ct]);
      }
    }
  }
  // epilogue: normalize and store
  #pragma unroll
  for (int ct = 0; ct < 4; ++ct) {
    #pragma unroll
    for (int r = 0; r < 8; ++r) {
      int row = qt * 16 + r + hi8;
      if (row < N_) {
        outb[(size_t)(b * N_ + row) * C_ + h * DH_ + ct * 16 + n16] = acc[ct][r] / lrun[r];
      }
    }
  }
}

// ---------------------------------------------------------------------------
// Combine: cls = 0.5*(pos+neg); body selected by keep_neg then keep_pos.
// ---------------------------------------------------------------------------
__global__ __launch_bounds__(256) void combine_kernel(
    const float* __restrict__ op, const float* __restrict__ on,
    const float* __restrict__ keepp, const float* __restrict__ keepn,
    float* __restrict__ xo)
{
  size_t i = (size_t)blockIdx.x * blockDim.x + threadIdx.x;
  if (i >= (size_t)M_ROWS * C_) return;
  int row = (int)(i / C_);
  int n = row % N_;
  float vp = op[i], vn = on[i];
  float r;
  if (n == 0) r = 0.5f * (vp + vn);
  else {
    int t = n - 1;
    r = (keepn[t] > 0.5f) ? vn : ((keepp[t] > 0.5f) ? vp : 0.f);
  }
  xo[i] = r;
}

__global__ void tail_kernel(const float* __restrict__ g_info, float* __restrict__ dst)
{
  int i = blockIdx.x * blockDim.x + threadIdx.x;
  if (i < 2 * C_) dst[i] = g_info[2 * C_ + i];   // g_info[1:] flat
}

// ---------------------------------------------------------------------------
extern "C" void kernel_launch(void* const* d_in, const int* in_sizes, int n_in,
                              void* d_out, int out_size, void* d_ws, size_t ws_size,
                              hipStream_t stream)
{
  const float* x      = (const float*)d_in[0];
  const float* g_info = (const float*)d_in[1];
  const float* w_qkv  = (const float*)d_in[2];
  const float* w_proj = (const float*)d_in[3];
  const float* b_proj = (const float*)d_in[4];
  float* out = (float*)d_out;

  float* ws   = (float*)d_ws;
  float* qkv  = ws;                                    // 4100*2304
  float* qsim = qkv  + (size_t)M_ROWS * QKV_COLS;      // 4096
  float* mpos = qsim + B_ * NT_;                       // 4096
  float* mneg = mpos + B_ * NT_;                       // 4096
  float* keepp= mneg + B_ * NT_;                       // 1024
  float* keepn= keepp + NT_;                           // 1024
  float* outp = keepn + NT_;                           // 4100*768
  float* outn = outp + (size_t)M_ROWS * C_;            // 4100*768
  float* xo   = qkv;                                   // reuse qkv (dead after attention)

  const int tilesM = (M_ROWS + 15) / 16;               // 257

  // 1) qkv = x @ w_qkv          (4100x768)*(768x2304)
  gemm_async_kernel<<<dim3(tilesM, QKV_COLS / GEMM_BN), 128, 0, stream>>>(
      x, w_qkv, nullptr, qkv, M_ROWS, QKV_COLS, C_);
  // 2) q_sim per (b, token)
  qsim_kernel<<<(B_ * NT_ * 32 + 255) / 256, 256, 0, stream>>>(qkv, g_info, qsim);
  // 3) min/max normalize + masks
  mask_kernel<<<1, 1024, 0, stream>>>(qsim, mpos, mneg, keepp, keepn);
  // 4) two-group masked attention
  attn_kernel<<<dim3(65, B_ * H_, 2), 32, 0, stream>>>(qkv, mpos, mneg, keepp, keepn,
                                                       outp, outn);
  // 5) combine into xo
  {
    size_t total = (size_t)M_ROWS * C_;
    combine_kernel<<<(unsigned)((total + 255) / 256), 256, 0, stream>>>(outp, outn,
                                                                        keepp, keepn, xo);
  }
  // 6) out = xo @ w_proj + b_proj   (4100x768)*(768x768)
  gemm_async_kernel<<<dim3(tilesM, C_ / GEMM_BN), 128, 0, stream>>>(
      xo, w_proj, b_proj, out, M_ROWS, C_, C_);
  // 7) second output: g_info[1:]
  tail_kernel<<<(2 * C_ + 255) / 256, 256, 0, stream>>>(g_info, out + (size_t)M_ROWS * C_);
}